// FusedSpeculativeBase_73564199846327
// MI455X (gfx1250) — compile-verified
//
#include <hip/hip_runtime.h>

// FusedSpeculativeBase: pad-replace + per-row truncate-to-count, plus counts
// pass-through copy. Memory-bound: ~131 MB total traffic -> ~5.6 us at 23.3 TB/s.
//
// Layout:
//   d_in[0]: output_tokens  int32 [B*K]   (B=262144, K=64)
//   d_in[1]: output_counts  int32 [B]
//   d_in[2]: pad_token_id   int32 [1]
//   d_out  : result_tokens  int32 [B*K]  ++  output_counts int32 [B]
//
// Each thread processes one int4 (4 tokens) -> 128-bit global load/store.
// K/4 = 16 int4 per row; a wave32 covers exactly 2 rows, fully coalesced.
// Tail threads (beyond the token range) copy output_counts as int4.

typedef int v4i __attribute__((ext_vector_type(4)));

__global__ __launch_bounds__(256) void spec_pad_trunc_kernel(
    const int* __restrict__ tokens,   // [B*K]
    const int* __restrict__ counts,   // [B]
    const int* __restrict__ pad_ptr,  // [1]
    int* __restrict__ out_tokens,     // [B*K]
    int* __restrict__ out_counts,     // [B]
    int n_tok4,                       // B*K/4
    int n_cnt4)                       // B/4
{
    const int idx = blockIdx.x * blockDim.x + threadIdx.x;

    // Uniform scalars: pad id and the pre-replacement tokens[0][0].
    // Force into SGPRs so the inner select is v_cmp + v_cndmask vs SGPR.
    int pad = __builtin_amdgcn_readfirstlane(pad_ptr[0]);
    int np  = __builtin_amdgcn_readfirstlane(tokens[0]);

    if (idx < n_tok4) {
        // Streamed once: non-temporal 128-bit load.
        const v4i* __restrict__ t4 = (const v4i*)tokens;
        v4i v = __builtin_nontemporal_load(t4 + idx);

        const int row = idx >> 4;          // K/4 == 16 int4 per row
        const int col = (idx & 15) << 2;   // first token column of this int4
        const int cnt = counts[row];       // 16 lanes share; L0/L2 broadcast

        v4i o;
        o.x = (col + 0 < cnt) ? ((v.x == pad) ? np : v.x) : pad;
        o.y = (col + 1 < cnt) ? ((v.y == pad) ? np : v.y) : pad;
        o.z = (col + 2 < cnt) ? ((v.z == pad) ? np : v.z) : pad;
        o.w = (col + 3 < cnt) ? ((v.w == pad) ? np : v.w) : pad;

        __builtin_nontemporal_store(o, (v4i*)out_tokens + idx);
    } else {
        // Tail: vectorized pass-through copy of output_counts.
        const int j = idx - n_tok4;
        if (j < n_cnt4) {
            v4i c = __builtin_nontemporal_load((const v4i*)counts + j);
            __builtin_nontemporal_store(c, (v4i*)out_counts + j);
        }
    }
}

extern "C" void kernel_launch(void* const* d_in, const int* in_sizes, int n_in,
                              void* d_out, int out_size, void* d_ws, size_t ws_size,
                              hipStream_t stream) {
    const int* tokens  = (const int*)d_in[0];
    const int* counts  = (const int*)d_in[1];
    const int* pad_ptr = (const int*)d_in[2];

    const int BK = in_sizes[0];   // B*K = 16,777,216
    const int B  = in_sizes[1];   // 262,144

    int* out_tokens = (int*)d_out;
    int* out_counts = (int*)d_out + BK;

    const int n_tok4 = BK >> 2;   // 4,194,304 int4
    const int n_cnt4 = B  >> 2;   // 65,536 int4

    const int total   = n_tok4 + n_cnt4;
    const int threads = 256;                       // 8 wave32 per block
    const int blocks  = (total + threads - 1) / threads;

    spec_pad_trunc_kernel<<<blocks, threads, 0, stream>>>(
        tokens, counts, pad_ptr, out_tokens, out_counts, n_tok4, n_cnt4);
}